// TSPNNet_47966194762329
// MI455X (gfx1250) — compile-verified
//
#include <hip/hip_runtime.h>
#include <hip/hip_bf16.h>

#define B_DIM 4
#define N_DIM 512
#define IN_DIM 4
#define E_DIM 128
#define EDGE_DIM 16
#define C_DIM 128
#define L_DIM 3
#define LN_EPS 1e-5f
#define DEG_EPS 1e-6f

// K-panel depth staged in LDS for the bmm kernel
#define KB 32

// Use gfx1250 async global->LDS copies (ASYNCcnt path). Set to 0 to fall back
// to plain vector load/store staging if the assembler rejects the mnemonic.
#define USE_ASYNC_LDS 1

typedef __attribute__((ext_vector_type(2))) float v2f;
typedef __attribute__((ext_vector_type(4))) float v4f;
typedef __attribute__((ext_vector_type(8))) float v8f;

// ---------------------------------------------------------------------------
// Kernel 1: collapse the edge-gating path to two scalars.
//   alpha = sum_k W_ee[k,0] * (sum_e W_ep[e,k])
//   beta  = sum_k b_ee[k]   * (sum_e W_ep[e,k]) + sum_e b_ep[e]
// ---------------------------------------------------------------------------
__global__ void scalars_kernel(const float* __restrict__ W_ee,
                               const float* __restrict__ b_ee,
                               const float* __restrict__ W_ep,
                               const float* __restrict__ b_ep,
                               float* __restrict__ scal) {
    __shared__ float colsum[EDGE_DIM];
    int t = threadIdx.x;
    if (t < EDGE_DIM) {
        float s = 0.f;
        for (int e = 0; e < E_DIM; ++e) s += W_ep[e * EDGE_DIM + t];
        colsum[t] = s;
    }
    __syncthreads();
    if (t == 0) {
        float alpha = 0.f, beta = 0.f;
        for (int k = 0; k < EDGE_DIM; ++k) {
            alpha += W_ee[k] * colsum[k];
            beta  += b_ee[k] * colsum[k];
        }
        float sb = 0.f;
        for (int e = 0; e < E_DIM; ++e) sb += b_ep[e];
        scal[0] = alpha;
        scal[1] = beta + sb;
    }
}

// ---------------------------------------------------------------------------
// Kernel 2: node embedding  x0 = nf @ W_emb^T + b_emb   (K=4, trivial)
// ---------------------------------------------------------------------------
__global__ void embed_kernel(const float* __restrict__ nf,
                             const float* __restrict__ W_emb,
                             const float* __restrict__ b_emb,
                             float* __restrict__ xinit) {
    int row = blockIdx.x;
    int e = threadIdx.x;
    const float* nrow = nf + (size_t)row * IN_DIM;
    float acc = b_emb[e];
#pragma unroll
    for (int j = 0; j < IN_DIM; ++j) acc += nrow[j] * W_emb[e * IN_DIM + j];
    xinit[(size_t)row * C_DIM + e] = acc;
}

// ---------------------------------------------------------------------------
// Kernel 3: inverse degree per row.  block per row (256 threads)
// ---------------------------------------------------------------------------
__global__ void invdeg_kernel(const float* __restrict__ adj,
                              float* __restrict__ invdeg) {
    __shared__ float red[256];
    int row = blockIdx.x;  // b*N + n
    const float* r = adj + (size_t)row * N_DIM;
    float s = 0.f;
    for (int m = threadIdx.x; m < N_DIM; m += 256) s += r[m];
    red[threadIdx.x] = s;
    __syncthreads();
    for (int off = 128; off > 0; off >>= 1) {
        if (threadIdx.x < (unsigned)off) red[threadIdx.x] += red[threadIdx.x + off];
        __syncthreads();
    }
    if (threadIdx.x == 0) invdeg[row] = 1.0f / (red[0] + DEG_EPS);
}

// ---------------------------------------------------------------------------
// Kernel 4: A = (adj * invdeg_row) * (alpha*adj + beta)   elementwise
// ---------------------------------------------------------------------------
__global__ void build_A_kernel(const float* __restrict__ adj,
                               const float* __restrict__ invdeg,
                               const float* __restrict__ scal,
                               float* __restrict__ Anorm) {
    size_t idx = (size_t)blockIdx.x * blockDim.x + threadIdx.x;
    float alpha = scal[0], beta = scal[1];
    float a = adj[idx];
    size_t row = idx / N_DIM;  // b*N + n
    Anorm[idx] = a * invdeg[row] * (alpha * a + beta);
}

// ---------------------------------------------------------------------------
// Kernel 5: batched agg = A[b] (512x512) @ X[b] (512x128).
// Block = 256 threads (8 waves); wave w owns rows [row0, row0+16) and ALL
// 128 output columns (8 v8f accumulators).  X is staged per-block into LDS
// in KB-row panels via gfx1250 async global->LDS copies, shared by all
// waves.  Each A fragment is loaded once and reused across the 8 column
// tiles, feeding 8 back-to-back v_wmma_f32_16x16x4_f32.
//
// Fragment layouts (ISA 7.12.2, wave32):
//   A 16x4 f32 : lanes0-15 = M, vgpr0/1 = K0/K1; lanes16-31 = K2/K3
//   B 4x16 f32 : lane&15 = N, lane half picks K pair, vgpr = K within pair
//   D 16x16 f32: vgpr r: lanes0-15 -> M=r, lanes16-31 -> M=8+r, lane&15 = N
// ---------------------------------------------------------------------------
__global__ void bmm_wmma_kernel(const float* __restrict__ A,
                                const float* __restrict__ X,
                                float* __restrict__ Y) {
    __shared__ float Xs[KB * C_DIM];  // 16 KB
    const int blocksPerBatch = N_DIM / 128;  // 4
    int wave = threadIdx.x >> 5;
    int lane = threadIdx.x & 31;
    int b   = blockIdx.x / blocksPerBatch;
    int rb  = blockIdx.x % blocksPerBatch;
    int row0 = rb * 128 + wave * 16;
    const float* Ab = A + (size_t)b * N_DIM * N_DIM + (size_t)row0 * N_DIM;
    const float* Xb = X + (size_t)b * N_DIM * C_DIM;
    int half = lane >> 4, l16 = lane & 15;

    v8f acc[8];
#pragma unroll
    for (int t = 0; t < 8; ++t) acc[t] = (v8f){};

    for (int k0 = 0; k0 < N_DIM; k0 += KB) {
        // ---- stage X[k0 .. k0+KB) x 128 into LDS (4096 floats) ----
        const float* src = Xb + (size_t)k0 * C_DIM;
#if USE_ASYNC_LDS
        {
            unsigned ldsbase = (unsigned)(size_t)(&Xs[0]);
#pragma unroll
            for (int j = 0; j < 4; ++j) {
                unsigned fid = threadIdx.x + j * 256;          // float4 index
                const float* g = src + (size_t)fid * 4;
                unsigned loff = ldsbase + fid * 16;
                asm volatile("global_load_async_to_lds_b128 %0, %1, off"
                             :: "v"(loff), "v"(g) : "memory");
            }
            // prefetch next K-panel while the async DMA runs
            if (k0 + KB < N_DIM)
                __builtin_prefetch(src + (size_t)KB * C_DIM + threadIdx.x * 16, 0, 1);
            asm volatile("s_wait_asynccnt 0" ::: "memory");
        }
#else
        {
#pragma unroll
            for (int j = 0; j < 4; ++j) {
                unsigned fid = threadIdx.x + j * 256;
                ((v4f*)Xs)[fid] = ((const v4f*)src)[fid];
            }
            if (k0 + KB < N_DIM)
                __builtin_prefetch(src + (size_t)KB * C_DIM + threadIdx.x * 16, 0, 1);
        }
#endif
        __syncthreads();

        // ---- compute: 8 K-steps x 8 column tiles = 64 WMMA per panel ----
#pragma unroll
        for (int kk = 0; kk < KB; kk += 4) {
            int ka = kk + 2 * half;
            const float* arow = Ab + (size_t)l16 * N_DIM + (k0 + ka);
            v2f af = { arow[0], arow[1] };
#pragma unroll
            for (int t = 0; t < 8; ++t) {
                v2f bf = { Xs[ka * C_DIM + t * 16 + l16],
                           Xs[(ka + 1) * C_DIM + t * 16 + l16] };
                acc[t] = __builtin_amdgcn_wmma_f32_16x16x4_f32(
                    false, af, false, bf, (short)0, acc[t], false, false);
            }
        }
        __syncthreads();
    }

    float* Yb = Y + (size_t)b * N_DIM * C_DIM + (size_t)row0 * C_DIM;
#pragma unroll
    for (int t = 0; t < 8; ++t) {
#pragma unroll
        for (int r = 0; r < 8; ++r) {
            int m = r + 8 * half;
            Yb[(size_t)m * C_DIM + t * 16 + l16] = acc[t][r];
        }
    }
}

// ---------------------------------------------------------------------------
// Kernel 6: out = agg @ W^T + b  -> LayerNorm -> + xinit -> ReLU, fused.
// Block = 256 threads (8 waves); each wave computes one 16x16 column tile of
// a 16-row x 128-col strip via WMMA into LDS; then LN over LDS rows with
// half-wave (16-lane) shuffle reductions.  grid = (B*N)/16 = 128.
// ---------------------------------------------------------------------------
__global__ void conv_ln_kernel(const float* __restrict__ Agg,
                               const float* __restrict__ Wc,   // (C,C) row-major
                               const float* __restrict__ bc,   // (C)
                               const float* __restrict__ g,    // (C)
                               const float* __restrict__ bln,  // (C)
                               const float* __restrict__ Xinit,
                               float* __restrict__ Xout) {
    __shared__ float tile[16 * C_DIM];  // 8 KB
    int wave = threadIdx.x >> 5;        // column tile 0..7
    int lane = threadIdx.x & 31;
    int row0 = blockIdx.x * 16;
    int half = lane >> 4, l16 = lane & 15;
    int col0 = wave * 16;

    const float* Ar = Agg + (size_t)row0 * C_DIM;
    v8f acc = {};
#pragma unroll 4
    for (int k0 = 0; k0 < C_DIM; k0 += 4) {
        int ka = k0 + 2 * half;
        // out[:,c] = sum_k agg[:,k] * W[c,k]  => B(k,c) = W[c,k]
        v2f afrag = { Ar[(size_t)l16 * C_DIM + ka], Ar[(size_t)l16 * C_DIM + ka + 1] };
        v2f bfrag = { Wc[(size_t)(col0 + l16) * C_DIM + ka],
                      Wc[(size_t)(col0 + l16) * C_DIM + ka + 1] };
        acc = __builtin_amdgcn_wmma_f32_16x16x4_f32(
            false, afrag, false, bfrag, (short)0, acc, false, false);
    }
#pragma unroll
    for (int r = 0; r < 8; ++r) {
        int m = r + 8 * half;
        tile[m * C_DIM + col0 + l16] = acc[r];
    }
    __syncthreads();

    // LayerNorm: 16 groups of 16 threads, one group per row.
    int rrow = threadIdx.x >> 4;  // 0..15
    int rl   = threadIdx.x & 15;  // 0..15
    float vals[8];
    float s = 0.f, q = 0.f;
#pragma unroll
    for (int j = 0; j < 8; ++j) {
        int c = rl + j * 16;
        float v = tile[rrow * C_DIM + c] + bc[c];
        vals[j] = v;
        s += v;
        q += v * v;
    }
#pragma unroll
    for (int mask = 8; mask > 0; mask >>= 1) {
        s += __shfl_xor(s, mask, 16);
        q += __shfl_xor(q, mask, 16);
    }
    float mean = s * (1.f / C_DIM);
    float var  = q * (1.f / C_DIM) - mean * mean;
    float rstd = rsqrtf(var + LN_EPS);
#pragma unroll
    for (int j = 0; j < 8; ++j) {
        int c = rl + j * 16;
        float o = (vals[j] - mean) * rstd * g[c] + bln[c]
                + Xinit[(size_t)(row0 + rrow) * C_DIM + c];
        Xout[(size_t)(row0 + rrow) * C_DIM + c] = fmaxf(o, 0.f);
    }
}

// ---------------------------------------------------------------------------
// Kernel 7: pooled = mean over N.  grid = B, block = 128 (one thread per c).
// ---------------------------------------------------------------------------
__global__ void pool_kernel(const float* __restrict__ X, float* __restrict__ pooled) {
    int b = blockIdx.x;
    int c = threadIdx.x;
    const float* Xb = X + (size_t)b * N_DIM * C_DIM;
    float s = 0.f;
    for (int n = 0; n < N_DIM; ++n) s += Xb[(size_t)n * C_DIM + c];
    pooled[b * C_DIM + c] = s * (1.f / N_DIM);
}

// ---------------------------------------------------------------------------
// Kernel 8: heads.  pi[b,n] = pooled[b]·W_pi[n] + b_pi[n];  v[b] = pooled[b]·W_v + b_v
// out layout: [pi (B*N) | v (B)]
// ---------------------------------------------------------------------------
__global__ void head_kernel(const float* __restrict__ pooled,
                            const float* __restrict__ W_pi,
                            const float* __restrict__ b_pi,
                            const float* __restrict__ W_v,
                            const float* __restrict__ b_v,
                            float* __restrict__ out) {
    int idx = blockIdx.x * blockDim.x + threadIdx.x;  // 0..B*N-1
    int b = idx / N_DIM, n = idx % N_DIM;
    const float* p = pooled + b * C_DIM;
    float s = b_pi[n];
#pragma unroll 4
    for (int c = 0; c < C_DIM; ++c) s += p[c] * W_pi[n * C_DIM + c];
    out[idx] = s;
    if (idx < B_DIM) {
        const float* pb = pooled + idx * C_DIM;
        float sv = b_v[0];
#pragma unroll 4
        for (int c = 0; c < C_DIM; ++c) sv += pb[c] * W_v[c];
        out[B_DIM * N_DIM + idx] = sv;
    }
}

// ---------------------------------------------------------------------------
extern "C" void kernel_launch(void* const* d_in, const int* in_sizes, int n_in,
                              void* d_out, int out_size, void* d_ws, size_t ws_size,
                              hipStream_t stream) {
    const float* node_features = (const float*)d_in[0];
    const float* adjacency     = (const float*)d_in[1];
    const float* W_emb         = (const float*)d_in[2];
    const float* b_emb         = (const float*)d_in[3];
    const float* W_ee          = (const float*)d_in[4];
    const float* b_ee          = (const float*)d_in[5];
    const float* W_ep          = (const float*)d_in[6];
    const float* b_ep          = (const float*)d_in[7];
    const float* W_conv        = (const float*)d_in[8];
    const float* b_conv        = (const float*)d_in[9];
    const float* ln_g          = (const float*)d_in[10];
    const float* ln_b          = (const float*)d_in[11];
    const float* W_pi          = (const float*)d_in[12];
    const float* b_pi          = (const float*)d_in[13];
    const float* W_v           = (const float*)d_in[14];
    const float* b_v           = (const float*)d_in[15];
    float* out = (float*)d_out;

    // workspace layout (floats)
    float* ws = (float*)d_ws;
    float* Anorm  = ws;                                      // B*N*N      = 1048576
    float* xinit  = Anorm + (size_t)B_DIM * N_DIM * N_DIM;   // B*N*C      =  262144
    float* xa     = xinit + (size_t)B_DIM * N_DIM * C_DIM;
    float* xb     = xa    + (size_t)B_DIM * N_DIM * C_DIM;
    float* agg    = xb    + (size_t)B_DIM * N_DIM * C_DIM;
    float* invdeg = agg   + (size_t)B_DIM * N_DIM * C_DIM;   // B*N        =    2048
    float* scal   = invdeg + (size_t)B_DIM * N_DIM;          // 2
    float* pooled = scal + 2;                                // B*C        =     512

    scalars_kernel<<<1, 32, 0, stream>>>(W_ee, b_ee, W_ep, b_ep, scal);
    embed_kernel<<<B_DIM * N_DIM, C_DIM, 0, stream>>>(node_features, W_emb, b_emb, xinit);
    invdeg_kernel<<<B_DIM * N_DIM, 256, 0, stream>>>(adjacency, invdeg);
    build_A_kernel<<<(B_DIM * N_DIM * N_DIM) / 256, 256, 0, stream>>>(adjacency, invdeg, scal, Anorm);

    const int bmmBlocks = B_DIM * (N_DIM / 128);  // 8 waves/block, 128 rows/block
    const float* xcur = xinit;
    float* xnext = xb;
    for (int i = 0; i < L_DIM; ++i) {
        bmm_wmma_kernel<<<bmmBlocks, 256, 0, stream>>>(Anorm, xcur, agg);
        conv_ln_kernel<<<(B_DIM * N_DIM) / 16, 256, 0, stream>>>(
            agg, W_conv + (size_t)i * C_DIM * C_DIM, b_conv + (size_t)i * C_DIM,
            ln_g + (size_t)i * C_DIM, ln_b + (size_t)i * C_DIM, xinit, xnext);
        xcur = xnext;
        xnext = (xnext == xb) ? xa : xb;
    }
    // after 3 layers: xcur == xb

    pool_kernel<<<B_DIM, C_DIM, 0, stream>>>(xcur, pooled);
    head_kernel<<<(B_DIM * N_DIM) / 256, 256, 0, stream>>>(pooled, W_pi, b_pi, W_v, b_v, out);
}